// HyperConvBlock_70480413327623
// MI455X (gfx1250) — compile-verified
//
#include <hip/hip_runtime.h>
#include <hip/hip_bf16.h>
#include <math.h>

// ---------------------------------------------------------------------------
// HyperConvBlock on MI455X (gfx1250, wave32).
// Six chained f32 WMMA GEMMs (V_WMMA_F32_16X16X4_F32). Problem is HBM-bound
// (~170 GFLOP vs ~200MB traffic -> ~9us floor @ 23.3 TB/s), so keep full f32
// precision and spend effort on data movement. 64x128 block tile, 8 waves of
// 32x32 outputs (4 independent WMMA accumulation chains/wave, 1 LDS b64 per
// WMMA), double-buffered LDS staging, one barrier per 32-deep K chunk.
// ---------------------------------------------------------------------------

typedef float v2f __attribute__((ext_vector_type(2)));
typedef float v4f __attribute__((ext_vector_type(4)));
typedef float v8f __attribute__((ext_vector_type(8)));

#define BM  64    // rows per workgroup tile
#define BN  128   // cols per workgroup tile
#define KC  32    // K chunk staged in LDS per double-buffer slot
#define AKP 36    // LDS pitch A tile: 144B = 9*16 (b128-store safe, bank-spread)
#define BKP 36    // LDS pitch B tile (transposed [n][k]): b64-frag safe

__device__ __forceinline__ v8f wmma4(v2f a, v2f b, v8f c) {
    return __builtin_amdgcn_wmma_f32_16x16x4_f32(
        false, a, false, b, (short)0, c, false, false);
}

// Epilogue modes:
//   0: relu(acc + biasCol[c])
//   1: acc + biasCol[c]
//   2: sin(acc + Emat[b,r,c])                      (bias matrix)
//   3: (acc + biasCol[c] + Emat[b,r,c]) * 0.5f     (residual + halve)
template <int MODE, int RPB>
__global__ __launch_bounds__(256)
void gemm_f32_wmma(const float* __restrict__ A, long bsA, long rsA,
                   const float* __restrict__ Bm, long bsB,
                   const float* __restrict__ biasCol,
                   const float* __restrict__ Emat, long bsE, long rsE,
                   float* __restrict__ Out,
                   int Mtotal, int K, int klim, int N)
{
    __shared__ float As[2][BM][AKP];
    __shared__ float Bs[2][BN][BKP];   // transposed: Bs[n][k]

    const int tid  = threadIdx.x;
    const int lane = tid & 31;
    const int wave = tid >> 5;
    const int wm   = wave & 1;   // 2 waves along M (32 rows each)
    const int wn   = wave >> 1;  // 4 waves along N (32 cols each)

    const int tileM = blockIdx.y * BM;
    const int tileN = blockIdx.x * BN;
    // batch for per-batch B operand (requires RPB % BM == 0 when bsB != 0)
    const int bB = (bsB != 0) ? (tileM / RPB) : 0;

    v8f acc00 = {}, acc01 = {}, acc10 = {}, acc11 = {};

    // --- A loader: thread -> (row = tid/4, two k-quads at kc, kc+16) -------
    const int a_row = tid >> 2;
    const int a_kc  = (tid & 3) << 2;
    int a_grow = tileM + a_row;
    if (a_grow >= Mtotal) a_grow = Mtotal - 1;          // clamp (never stored)
    const int a_b = a_grow / RPB;                        // compile-time RPB
    const int a_r = a_grow - a_b * RPB;
    const float* a_base = A + (long)a_b * bsA + (long)a_r * rsA;

    // --- B loader: thread -> rows {kr, kr+16} x n-quads {nc, nc+64} --------
    const int b_kr = tid >> 4;                           // 0..15
    const int b_nc = (tid & 15) << 2;                    // 0..60
    const float* b_base = Bm + (long)bB * bsB + tileN;

    // --- fragment coordinates (f32 WMMA layout: lane = m|n + 16*(k/2)) -----
    const int m0   = (wm << 5) + (lane & 15);            // rows m0, m0+16
    const int n0   = (wn << 5) + (lane & 15);            // cols n0, n0+16
    const int koff = (lane >> 4) << 1;                   // 0 | 2

    const int nchunks = (klim + KC - 1) / KC;

    v4f ra0, ra1, rb00, rb01, rb10, rb11;

    auto prefetch = [&](int k0) {
        if (k0 + KC <= klim) {                           // uniform fast path
            ra0  = *(const v4f*)(a_base + k0 + a_kc);
            ra1  = *(const v4f*)(a_base + k0 + a_kc + 16);
            const float* br0 = b_base + (long)(k0 + b_kr)      * N + b_nc;
            const float* br1 = b_base + (long)(k0 + b_kr + 16) * N + b_nc;
            rb00 = *(const v4f*)(br0);
            rb01 = *(const v4f*)(br0 + 64);
            rb10 = *(const v4f*)(br1);
            rb11 = *(const v4f*)(br1 + 64);
        } else {                                         // last partial chunk
            const v4f zz = {0.f, 0.f, 0.f, 0.f};
            #pragma unroll
            for (int i = 0; i < 4; ++i) {
                const int ka = k0 + a_kc + i;
                ra0[i] = (ka      < klim) ? a_base[ka]      : 0.0f;
                ra1[i] = (ka + 16 < klim) ? a_base[ka + 16] : 0.0f;
            }
            const int kb0 = k0 + b_kr, kb1 = kb0 + 16;
            const float* br0 = b_base + (long)kb0 * N + b_nc;
            const float* br1 = b_base + (long)kb1 * N + b_nc;
            rb00 = (kb0 < K) ? *(const v4f*)(br0)      : zz;
            rb01 = (kb0 < K) ? *(const v4f*)(br0 + 64) : zz;
            rb10 = (kb1 < K) ? *(const v4f*)(br1)      : zz;
            rb11 = (kb1 < K) ? *(const v4f*)(br1 + 64) : zz;
        }
    };

    auto stage = [&](int buf) {
        *(v4f*)&As[buf][a_row][a_kc]      = ra0;         // ds_store_b128
        *(v4f*)&As[buf][a_row][a_kc + 16] = ra1;
        #pragma unroll
        for (int i = 0; i < 4; ++i) {                    // transpose B -> [n][k]
            Bs[buf][b_nc + i     ][b_kr]      = rb00[i]; // pairs merge into
            Bs[buf][b_nc + i     ][b_kr + 16] = rb10[i]; //   ds_store_2addr_b32
            Bs[buf][b_nc + i + 64][b_kr]      = rb01[i];
            Bs[buf][b_nc + i + 64][b_kr + 16] = rb11[i];
        }
    };

    auto compute = [&](int buf) {
        #pragma unroll
        for (int kk = 0; kk < KC; kk += 4) {
            v2f a0 = *(const v2f*)&As[buf][m0     ][kk + koff];    // ds b64
            v2f a1 = *(const v2f*)&As[buf][m0 + 16][kk + koff];
            v2f b0 = *(const v2f*)&Bs[buf][n0     ][kk + koff];
            v2f b1 = *(const v2f*)&Bs[buf][n0 + 16][kk + koff];
            acc00 = wmma4(a0, b0, acc00);                // 4 independent
            acc01 = wmma4(a0, b1, acc01);                //   accumulation
            acc10 = wmma4(a1, b0, acc10);                //   chains
            acc11 = wmma4(a1, b1, acc11);
        }
    };

    prefetch(0);
    stage(0);
    __syncthreads();
    int buf = 0;
    for (int c = 0; c < nchunks; ++c) {
        if (c + 1 < nchunks) prefetch((c + 1) * KC);     // overlap with compute
        compute(buf);
        if (c + 1 < nchunks) {
            stage(buf ^ 1);
            __syncthreads();
            buf ^= 1;
        }
    }

    // --- epilogue: C/D layout -> reg j = row j (lanes<16) / row j+8 --------
    const int colA  = tileN + n0;
    const int mbase = tileM + (wm << 5) + ((lane >> 4) << 3);
    #pragma unroll
    for (int mi = 0; mi < 2; ++mi) {
        const v8f& accL = mi ? acc10 : acc00;            // col n0
        const v8f& accR = mi ? acc11 : acc01;            // col n0+16
        #pragma unroll
        for (int j = 0; j < 8; ++j) {
            const int gr = mbase + (mi << 4) + j;
            if (gr >= Mtotal) continue;
            const int b = gr / RPB;                      // mul-shift (constexpr)
            const int r = gr - b * RPB;
            float v0 = accL[j];
            float v1 = accR[j];
            if (MODE == 0) {
                v0 = fmaxf(v0 + biasCol[colA],      0.0f);
                v1 = fmaxf(v1 + biasCol[colA + 16], 0.0f);
            } else if (MODE == 1) {
                v0 += biasCol[colA];
                v1 += biasCol[colA + 16];
            } else if (MODE == 2) {
                const float* e = Emat + (long)b * bsE + (long)r * rsE;
                v0 = sinf(v0 + e[colA]);
                v1 = sinf(v1 + e[colA + 16]);
            } else { // MODE == 3
                const float* e = Emat + (long)b * bsE + (long)r * rsE;
                v0 = (v0 + biasCol[colA]      + e[colA])      * 0.5f;
                v1 = (v1 + biasCol[colA + 16] + e[colA + 16]) * 0.5f;
            }
            Out[(long)gr * N + colA]      = v0;
            Out[(long)gr * N + colA + 16] = v1;
        }
    }
}

// ---------------------------------------------------------------------------
extern "C" void kernel_launch(void* const* d_in, const int* in_sizes, int n_in,
                              void* d_out, int out_size, void* d_ws, size_t ws_size,
                              hipStream_t stream)
{
    (void)in_sizes; (void)n_in; (void)out_size; (void)ws_size;

    const float* x  = (const float*)d_in[0];   // (8, 1024, 1024)
    const float* z  = (const float*)d_in[1];   // (8, 1028, 128)
    const float* W1 = (const float*)d_in[2];   // (3, 128, 1024)
    const float* b1 = (const float*)d_in[3];
    const float* W2 = (const float*)d_in[4];   // (3, 1024, 1024)
    const float* b2 = (const float*)d_in[5];
    const float* V1 = (const float*)d_in[6];
    const float* c1 = (const float*)d_in[7];
    const float* V2 = (const float*)d_in[8];
    const float* c2 = (const float*)d_in[9];
    const float* Wr = (const float*)d_in[10];  // (1024, 1024)
    const float* br = (const float*)d_in[11];
    float* out = (float*)d_out;                // (8, 1024, 1024)

    const int Bb = 8, T = 1024, F = 1024, ZC = 128, ZL = 1028, HL = 1026;

    // workspace layout (floats): h1 | g1 | wts | bmat | s  (~168 MB total)
    float* ws   = (float*)d_ws;
    const size_t H1N = (size_t)Bb * HL * F;
    const size_t WTN = (size_t)Bb * T * F;
    float* h1   = ws;
    float* g1   = h1 + H1N;
    float* wts  = g1 + H1N;
    float* bmat = wts + WTN;
    float* sm   = bmat + WTN;

    const dim3 blk(256);
    const dim3 gridA(F / BN, (Bb * HL + BM - 1) / BM);  // M = 8208
    const dim3 gridB(F / BN, (Bb * T) / BM);            // M = 8192

    // 1) h1 = relu(unfold(z) @ W1 + b1)   : K = 3*128 = 384
    gemm_f32_wmma<0, 1026><<<gridA, blk, 0, stream>>>(
        z, (long)ZL * ZC, ZC, W1, 0, b1, nullptr, 0, 0,
        h1, Bb * HL, 3 * ZC, 3 * ZC, F);
    // 2) g1 = relu(unfold(z) @ V1 + c1)
    gemm_f32_wmma<0, 1026><<<gridA, blk, 0, stream>>>(
        z, (long)ZL * ZC, ZC, V1, 0, c1, nullptr, 0, 0,
        g1, Bb * HL, 3 * ZC, 3 * ZC, F);
    // 3) wts = unfold(h1) @ W2 + b2       : K = 3*1024 = 3072
    gemm_f32_wmma<1, 1024><<<gridB, blk, 0, stream>>>(
        h1, (long)HL * F, F, W2, 0, b2, nullptr, 0, 0,
        wts, Bb * T, 3 * F, 3 * F, F);
    // 4) bmat = unfold(g1) @ V2 + c2
    gemm_f32_wmma<1, 1024><<<gridB, blk, 0, stream>>>(
        g1, (long)HL * F, F, V2, 0, c2, nullptr, 0, 0,
        bmat, Bb * T, 3 * F, 3 * F, F);
    // 5) s = sin( shift4(x) @ wts[b] + bmat ) : per-batch B, A masked @ k>=1020
    gemm_f32_wmma<2, 1024><<<gridB, blk, 0, stream>>>(
        x + 4, (long)T * F, T, wts, (long)T * F, nullptr,
        bmat, (long)T * F, F,
        sm, Bb * T, T, T - 4, F);
    // 6) out = (s @ Wr + br + x) / 2
    gemm_f32_wmma<3, 1024><<<gridB, blk, 0, stream>>>(
        sm, (long)T * F, F, Wr, 0, br,
        x, (long)T * F, F,
        out, Bb * T, F, F, F);
}